// CIN_84567906058657
// MI455X (gfx1250) — compile-verified
//
#include <hip/hip_runtime.h>
#include <math.h>
#include <stdint.h>

typedef float v2f __attribute__((ext_vector_type(2)));
typedef float v8f __attribute__((ext_vector_type(8)));

#define B_BATCH 512
#define F_FEAT  32
#define D_DIM   64
#define O_UNITS 128
#define SUM_UNITS 384

// ---------------------------------------------------------------------------
// Transpose W (O x K, row-major) -> Wt (K x O) so WMMA B-fragment loads are
// contiguous 64B half-wave segments. W is tiny (<= 2MB) and L2-resident.
// ---------------------------------------------------------------------------
__global__ void wt_transpose_kernel(const float* __restrict__ W,
                                    float* __restrict__ Wt, int K) {
    int idx = blockIdx.x * blockDim.x + threadIdx.x;
    if (idx >= K * O_UNITS) return;
    int k = idx >> 7;       // / 128
    int o = idx & 127;
    Wt[idx] = W[o * K + k];
}

// ---------------------------------------------------------------------------
// One CIN layer: y[b,o,d] = sum_{f,h} x[b,f,d]*hprev[b,h,d]*W[o,f*H+h] + b[o]
// GEMM with M = d-rows (per batch), N = 128 outputs, K = 32*H, A = z built
// on the fly in registers via V_WMMA_F32_16X16X4_F32 (fp32-exact path).
// One block per batch b; 8 waves; each wave owns a 16(d) x 64(o) tile.
// LDS staging uses gfx1250 async global->LDS copies (ASYNCcnt tracked).
//   x     : (B, 32, 64)  row-major
//   hprev : (B, 64, H)   row-major (ignored when FIRST, h == x)
//   Wt    : (K, 128)
//   y     : (B, 64, 128) row-major (d-major -> coalesced stores & reduction)
// ---------------------------------------------------------------------------
template <int H, bool FIRST>
__global__ __launch_bounds__(256)
void cin_layer_kernel(const float* __restrict__ x,
                      const float* __restrict__ hprev,
                      const float* __restrict__ Wt,
                      const float* __restrict__ bias,
                      float* __restrict__ y) {
    constexpr int HPAD = H + 4;                       // LDS bank-conflict padding
    __shared__ float x_s[F_FEAT * D_DIM];             // (f, d)
    __shared__ float h_s[FIRST ? 1 : D_DIM * HPAD];   // (d, h) padded

    const int tid = threadIdx.x;
    const int b   = blockIdx.x;

    // --- Async stage x[b]: 8KB, global_load_async_to_lds_b128 (GVS mode) ---
    {
        const char* xg = (const char*)(x + (size_t)b * F_FEAT * D_DIM);
        for (int i = tid; i < (F_FEAT * D_DIM) / 4; i += 256) {
            unsigned lds = (unsigned)(uintptr_t)&x_s[i * 4];
            unsigned off = (unsigned)i * 16u;
            asm volatile("global_load_async_to_lds_b128 %0, %1, %2"
                         :: "v"(lds), "v"(off), "s"(xg) : "memory");
        }
    }
    // --- Async stage hprev[b]: 32KB into padded (d, h) layout ---
    if (!FIRST) {
        const char* hg = (const char*)(hprev + (size_t)b * D_DIM * H);
        for (int i = tid; i < (D_DIM * H) / 4; i += 256) {
            int d  = (i * 4) / H;
            int hh = (i * 4) % H;
            unsigned lds = (unsigned)(uintptr_t)&h_s[d * HPAD + hh];
            unsigned off = (unsigned)i * 16u;
            asm volatile("global_load_async_to_lds_b128 %0, %1, %2"
                         :: "v"(lds), "v"(off), "s"(hg) : "memory");
        }
    }
    asm volatile("s_wait_asynccnt 0x0" ::: "memory");
    __syncthreads();

    const int lane = tid & 31;
    const int wave = tid >> 5;
    const int d0   = (wave & 3) * 16;     // 16-row d tile
    const int o0   = (wave >> 2) * 64;    // 64-col o half
    const int m    = lane & 15;           // row within tile (A/B/C: N or M = lane%16)
    const int kg   = (lane >> 4) << 1;    // K offset: lanes 0-15 -> 0, 16-31 -> 2

    // Seed accumulators with the bias (constant along M) -> pure-store epilogue.
    v8f acc[4];
    #pragma unroll
    for (int nt = 0; nt < 4; ++nt) {
        float bv = bias[o0 + nt * 16 + m];
        acc[nt] = (v8f){bv, bv, bv, bv, bv, bv, bv, bv};
    }

    const float* hr = FIRST ? x_s : &h_s[(d0 + m) * HPAD];

    for (int f = 0; f < F_FEAT; ++f) {
        const float xa = x_s[f * D_DIM + d0 + m];     // hoisted: 1 load / 32 k-steps
        // Near-cache prefetch of the next f-panel of Wt (wraps at the end).
        __builtin_prefetch(Wt + (size_t)((f + 1) & 31) * H * O_UNITS +
                           (size_t)lane * 32, 0, 3);
        #pragma unroll 4
        for (int h0 = 0; h0 < H; h0 += 4) {
            const int k = f * H + h0 + kg;
            // A fragment (16x4 f32): lane holds z[m, k] and z[m, k+1]
            v2f a;
            if (FIRST) {                  // layer 0: h == x, (f, d) layout
                a.x = xa * x_s[(h0 + kg) * D_DIM + d0 + m];
                a.y = xa * x_s[(h0 + kg + 1) * D_DIM + d0 + m];
            } else {                      // (d, h) padded layout, 8B-aligned pair
                float2 hp = *(const float2*)&hr[h0 + kg];
                a.x = xa * hp.x;
                a.y = xa * hp.y;
            }
            // B fragments (4x16 f32): VGPR0=K(k), VGPR1=K(k+1), N=lane%16
            const float* wrow = Wt + (size_t)k * O_UNITS + o0 + m;
            #pragma unroll
            for (int nt = 0; nt < 4; ++nt) {
                v2f bf;
                bf.x = wrow[nt * 16];
                bf.y = wrow[nt * 16 + O_UNITS];
                acc[nt] = __builtin_amdgcn_wmma_f32_16x16x4_f32(
                    false, a, false, bf, (short)0, acc[nt], false, false);
            }
        }
    }

    // Epilogue: C layout is VGPR r -> M = r (+8 for lanes 16-31), N = lane%16.
    const int dbase = d0 + ((lane >> 4) << 3);
    float* yb = y + (size_t)b * D_DIM * O_UNITS;
    #pragma unroll
    for (int nt = 0; nt < 4; ++nt) {
        const int o = o0 + nt * 16 + m;
        #pragma unroll
        for (int r = 0; r < 8; ++r)
            yb[(size_t)(dbase + r) * O_UNITS + o] = acc[nt][r];
    }
}

// outs[b, off + o] = sum_d y[b, d, o]   (coalesced across o each d step)
__global__ void reduce_d_kernel(const float* __restrict__ y,
                                float* __restrict__ outs, int off) {
    int b = blockIdx.x;
    int o = threadIdx.x;                  // 128 threads
    const float* yb = y + (size_t)b * D_DIM * O_UNITS + o;
    float s = 0.f;
    #pragma unroll 8
    for (int d = 0; d < D_DIM; ++d) s += yb[d * O_UNITS];
    outs[(size_t)b * SUM_UNITS + off + o] = s;
}

// out[b] = sigmoid( sum_c relu(outs[b,c]) * Wout[c] + bout )
__global__ void head_kernel(const float* __restrict__ outs,
                            const float* __restrict__ Wout,
                            const float* __restrict__ bout,
                            float* __restrict__ out) {
    int b = blockIdx.x * blockDim.x + threadIdx.x;
    if (b >= B_BATCH) return;
    const float* row = outs + (size_t)b * SUM_UNITS;
    float s = bout[0];
    for (int c = 0; c < SUM_UNITS; ++c)
        s += fmaxf(row[c], 0.f) * Wout[c];
    out[b] = 1.f / (1.f + __expf(-s));
}

// ---------------------------------------------------------------------------
// Workspace layout (floats):
//   wt0 (1024*128) | wt1 (4096*128) | wt2 (4096*128) |
//   yA (512*64*128) | yB (512*64*128) | outs (512*384)
// Total ~37.3 MB.
// ---------------------------------------------------------------------------
extern "C" void kernel_launch(void* const* d_in, const int* in_sizes, int n_in,
                              void* d_out, int out_size, void* d_ws, size_t ws_size,
                              hipStream_t stream) {
    const float* x    = (const float*)d_in[0];
    const float* W0   = (const float*)d_in[1];
    const float* b0   = (const float*)d_in[2];
    const float* W1   = (const float*)d_in[3];
    const float* b1   = (const float*)d_in[4];
    const float* W2   = (const float*)d_in[5];
    const float* b2   = (const float*)d_in[6];
    const float* Wout = (const float*)d_in[7];
    const float* bout = (const float*)d_in[8];
    float* out = (float*)d_out;

    float* ws   = (float*)d_ws;
    float* wt0  = ws;
    float* wt1  = wt0 + (size_t)1024 * 128;
    float* wt2  = wt1 + (size_t)4096 * 128;
    float* yA   = wt2 + (size_t)4096 * 128;
    float* yB   = yA  + (size_t)B_BATCH * D_DIM * O_UNITS;
    float* outs = yB  + (size_t)B_BATCH * D_DIM * O_UNITS;

    wt_transpose_kernel<<<(1024 * 128) / 256, 256, 0, stream>>>(W0, wt0, 1024);
    wt_transpose_kernel<<<(4096 * 128) / 256, 256, 0, stream>>>(W1, wt1, 4096);
    wt_transpose_kernel<<<(4096 * 128) / 256, 256, 0, stream>>>(W2, wt2, 4096);

    cin_layer_kernel<32,  true ><<<B_BATCH, 256, 0, stream>>>(x, nullptr, wt0, b0, yA);
    reduce_d_kernel<<<B_BATCH, 128, 0, stream>>>(yA, outs, 0);

    cin_layer_kernel<128, false><<<B_BATCH, 256, 0, stream>>>(x, yA, wt1, b1, yB);
    reduce_d_kernel<<<B_BATCH, 128, 0, stream>>>(yB, outs, 128);

    cin_layer_kernel<128, false><<<B_BATCH, 256, 0, stream>>>(x, yB, wt2, b2, yA);
    reduce_d_kernel<<<B_BATCH, 128, 0, stream>>>(yA, outs, 256);

    head_kernel<<<2, 256, 0, stream>>>(outs, Wout, bout, out);
}